// DynamicSuperLoss_25503515804375
// MI455X (gfx1250) — compile-verified
//
#include <hip/hip_runtime.h>
#include <hip/hip_bf16.h>
#include <float.h>

// ---------------------------------------------------------------------------
// DynamicSuperLoss for MI455X (gfx1250, wave32).
// Stage 1 (dominant): streaming row softmax-CE over 32768x1000 fp32 (~131 MB
// read -> ~5.6us HBM roofline) using CDNA5 async global->LDS staging with a
// double-buffered ASYNCcnt pipeline. Stage 2: tiny segment-EMA + 31-entry
// table interp + scalar mean.
// ---------------------------------------------------------------------------

#define WAVE 32
#define LN_SMOOTH (-0.1053605157f) /* ln(0.9), both branches use 0.9 */
#define LOG_OFFSET 1.3369849f      /* _LOG[0] + step*(N-1)/2        */
#define INV_LOG_STEP (1.0f / 5.5707704f)

#if defined(__has_builtin)
#if __has_builtin(__builtin_amdgcn_global_load_async_to_lds_b128)
#define HAS_ASYNC_LDS 1
#endif
#if __has_builtin(__builtin_amdgcn_s_wait_asynccnt)
#define HAS_WAIT_ASYNC 1
#endif
#endif

// Prototype of the async builtin (confirmed by round-2 diagnostic):
//   void __builtin_amdgcn_global_load_async_to_lds_b128(v4i AS1*, v4i AS3*,
//                                                       imm int off, imm int cpol)
typedef int v4i __attribute__((ext_vector_type(4)));
typedef __attribute__((address_space(1))) v4i* g_v4i_ptr;
typedef __attribute__((address_space(3))) v4i* l_v4i_ptr;

__constant__ float c_conf[31] = {
    0.9991138577461243f,  0.8724386692047119f,  0.8048540353775024f,
    0.7398145198822021f,  0.6715637445449829f,  0.5973713397979736f,
    0.5154045820236206f,  0.42423248291015625f, 0.3226756751537323f,
    0.20976418256759644f, 0.08473344892263412f, -0.05296758562326431f,
    -0.2036692053079605f, -0.3674810528755188f, -0.5443023443222046f,
    -0.7338425517082214f, -0.9356498718261719f, -1.149145483970642f,
    -1.3736592531204224f, -1.6084641218185425f, -1.8528070449829102f,
    -2.1059343814849854f, -2.367111921310425f,  -2.6356399059295654f,
    -2.910861015319824f,  -3.1921679973602295f, -3.479003667831421f,
    -3.770861864089966f,  -4.067285060882568f,  -4.367861747741699f,
    -4.67222261428833f};

__device__ __forceinline__ float optimal_conf(float l) {
    // emulate grid_sample 1D border-padded linear interp in log space
    float ln = (__logf(l + 0.750256f) - LOG_OFFSET) * INV_LOG_STEP;
    if (__builtin_isnan(ln)) ln = -1.0f;                 // log(neg) -> left border
    float ix = fminf(fmaxf((ln + 1.0f) * 15.0f, 0.0f), 30.0f);
    float i0f = floorf(ix);
    int   i0  = (int)i0f;
    float fr  = ix - i0f;
    int   i1  = (i0 < 30) ? i0 + 1 : 30;
    float r = c_conf[i0] * (1.0f - fr) + c_conf[i1] * fr;
    return __expf(r);
}

// Issue an async HBM->LDS copy of one row chunk (16B per active lane).
__device__ __forceinline__ void stage_row_async(const float* __restrict__ gsrc,
                                                float* __restrict__ lds_dst,
                                                int nvec, int t, int nthreads) {
#if defined(HAS_ASYNC_LDS)
    for (int j = t; j < nvec; j += nthreads) {
        __builtin_amdgcn_global_load_async_to_lds_b128(
            (g_v4i_ptr)(unsigned long long)(const void*)(gsrc + 4 * (size_t)j),
            (l_v4i_ptr)(unsigned int)(unsigned long long)(const void*)(lds_dst + 4 * (size_t)j),
            0, 0);
    }
#else
    for (int j = t; j < nvec; j += nthreads) {
        float4 v = ((const float4*)gsrc)[j];
        ((float4*)lds_dst)[j] = v;   // synchronous fallback
    }
#endif
}

__device__ __forceinline__ void wait_async0() {
#if defined(HAS_ASYNC_LDS)
#if defined(HAS_WAIT_ASYNC)
    __builtin_amdgcn_s_wait_asynccnt(0);
#else
    asm volatile("s_wait_asynccnt 0x0" ::: "memory");
#endif
#endif
}

// K0: zero only bins that this batch touches + class bins + output scalar.
__global__ void dsl_init(const int* __restrict__ indices, float* __restrict__ ins_sum,
                         float* __restrict__ ins_cnt, float* __restrict__ cls_sum,
                         float* __restrict__ cls_cnt, float* __restrict__ out,
                         int B, int ncls) {
    int t = blockIdx.x * blockDim.x + threadIdx.x;
    if (t < B) {
        int ix = indices[t];       // duplicates all write 0: benign race
        ins_sum[ix] = 0.0f;
        ins_cnt[ix] = 0.0f;
    }
    if (t < ncls) {
        cls_sum[t] = 0.0f;
        cls_cnt[t] = 0.0f;
    }
    if (t == 0) out[0] = 0.0f;
}

// K1 (async path): per-row online-softmax CE with double-buffered
// global_load_async_to_lds_b128 staging. Buffer for row i is buf[i&1].
// Protocol per iteration:
//   s_wait_asynccnt 0   -> my chunks of row i have landed in LDS
//   __syncthreads()     -> all waves landed row i AND finished reading the
//                          buffer that row i+1 will overwrite
//   issue row i+1 async -> overlaps with the exp/max work below
//   compute row i from LDS (ds_load_b128), label logit gathered from LDS
__global__ void dsl_ce_lds(const float* __restrict__ preds, const int* __restrict__ labels,
                           float* __restrict__ loss, int B, int ncls) {
    extern __shared__ float buf[];           // 2 * ncls floats
    __shared__ float sm[8];
    __shared__ float ss[8];
    const int nvec = ncls >> 2;
    const int t    = threadIdx.x;

    int row0 = blockIdx.x;
    if (row0 < B)
        stage_row_async(preds + (size_t)row0 * ncls, buf, nvec, t, blockDim.x);

    int iter = 0;
    for (int row = row0; row < B; row += gridDim.x, ++iter) {
        float* cur = buf + (size_t)(iter & 1) * ncls;
        float* nxt = buf + (size_t)((iter + 1) & 1) * ncls;

        wait_async0();
        __syncthreads();

        int nrow = row + gridDim.x;
        if (nrow < B)
            stage_row_async(preds + (size_t)nrow * ncls, nxt, nvec, t, blockDim.x);

        // thread-local online (max, sum-exp) from LDS; -FLT_MAX identity keeps
        // idle lanes NaN-free through the shuffle tree.
        float m = -FLT_MAX, s = 0.0f;
        for (int j = t; j < nvec; j += blockDim.x) {
            float4 v  = ((const float4*)cur)[j];
            float  m2 = fmaxf(fmaxf(v.x, v.y), fmaxf(v.z, v.w));
            float  mn = fmaxf(m, m2);
            s = s * __expf(m - mn) + __expf(v.x - mn) + __expf(v.y - mn) +
                __expf(v.z - mn) + __expf(v.w - mn);
            m = mn;
        }

        // wave32 reduction
        for (int off = WAVE / 2; off > 0; off >>= 1) {
            float mo = __shfl_xor(m, off, WAVE);
            float so = __shfl_xor(s, off, WAVE);
            float mn = fmaxf(m, mo);
            s = s * __expf(m - mn) + so * __expf(mo - mn);
            m = mn;
        }
        int wid  = t >> 5;
        int lane = t & (WAVE - 1);
        if (lane == 0) { sm[wid] = m; ss[wid] = s; }
        __syncthreads();
        if (t == 0) {
            float fm = sm[0], fs = ss[0];
            int nw = blockDim.x >> 5;
            for (int w = 1; w < nw; ++w) {
                float mo = sm[w], so = ss[w];
                float mn = fmaxf(fm, mo);
                fs = fs * __expf(fm - mn) + so * __expf(mo - mn);
                fm = mn;
            }
            float xl = cur[labels[row]];         // label logit from LDS
            loss[row] = fm + __logf(fs) - xl;    // logsumexp - x[label]
        }
        // no trailing barrier needed: next iteration's wait+barrier protects
        // both sm/ss reuse and the LDS buffer overwrite
    }
}

// K1 (direct fallback, any ncls): register-resident online softmax + prefetch.
__global__ void dsl_ce(const float* __restrict__ preds, const int* __restrict__ labels,
                       float* __restrict__ loss, int B, int ncls) {
    __shared__ float sm[8];
    __shared__ float ss[8];
    const int nvec = ncls >> 2;
    const int tail = nvec << 2;

    for (int row = blockIdx.x; row < B; row += gridDim.x) {
        const float4* rp = (const float4*)(preds + (size_t)row * ncls);
        int nrow = row + gridDim.x;
        if (nrow < B) {
            const float4* np = (const float4*)(preds + (size_t)nrow * ncls);
            for (int j = threadIdx.x; j < nvec; j += blockDim.x)
                __builtin_prefetch((const void*)(np + j), 0, 0);
        }
        float m = -FLT_MAX, s = 0.0f;
        for (int j = threadIdx.x; j < nvec; j += blockDim.x) {
            float4 v  = rp[j];
            float  m2 = fmaxf(fmaxf(v.x, v.y), fmaxf(v.z, v.w));
            float  mn = fmaxf(m, m2);
            s = s * __expf(m - mn) + __expf(v.x - mn) + __expf(v.y - mn) +
                __expf(v.z - mn) + __expf(v.w - mn);
            m = mn;
        }
        for (int j = tail + threadIdx.x; j < ncls; j += blockDim.x) {
            float x  = preds[(size_t)row * ncls + j];
            float mn = fmaxf(m, x);
            s = s * __expf(m - mn) + __expf(x - mn);
            m = mn;
        }
        for (int off = WAVE / 2; off > 0; off >>= 1) {
            float mo = __shfl_xor(m, off, WAVE);
            float so = __shfl_xor(s, off, WAVE);
            float mn = fmaxf(m, mo);
            s = s * __expf(m - mn) + so * __expf(mo - mn);
            m = mn;
        }
        int wid  = threadIdx.x >> 5;
        int lane = threadIdx.x & (WAVE - 1);
        if (lane == 0) { sm[wid] = m; ss[wid] = s; }
        __syncthreads();
        if (threadIdx.x == 0) {
            float fm = sm[0], fs = ss[0];
            int nw = blockDim.x >> 5;
            for (int w = 1; w < nw; ++w) {
                float mo = sm[w], so = ss[w];
                float mn = fmaxf(fm, mo);
                fs = fs * __expf(fm - mn) + so * __expf(mo - mn);
                fm = mn;
            }
            float xl = preds[(size_t)row * ncls + labels[row]];
            loss[row] = fm + __logf(fs) - xl;
        }
        __syncthreads();
    }
}

// K2: per-bin sum/count scatter for both branches.
__global__ void dsl_scatter(const float* __restrict__ loss, const int* __restrict__ labels,
                            const int* __restrict__ indices, float* __restrict__ ins_sum,
                            float* __restrict__ ins_cnt, float* __restrict__ cls_sum,
                            float* __restrict__ cls_cnt, int B) {
    int t = blockIdx.x * blockDim.x + threadIdx.x;
    if (t >= B) return;
    float lb = loss[t];
    int   ix = indices[t];
    int   lc = labels[t];
    atomicAdd(&ins_sum[ix], lb);
    atomicAdd(&ins_cnt[ix], 1.0f);
    atomicAdd(&cls_sum[lc], lb);
    atomicAdd(&cls_cnt[lc], 1.0f);
}

// K3: per-sample EMA + table interp for both branches, then mean.
__global__ void dsl_finalize(const float* __restrict__ loss, const int* __restrict__ labels,
                             const int* __restrict__ indices,
                             const float* __restrict__ mem_ins,
                             const float* __restrict__ mem_cls,
                             const float* __restrict__ ins_sum,
                             const float* __restrict__ ins_cnt,
                             const float* __restrict__ cls_sum,
                             const float* __restrict__ cls_cnt,
                             float* __restrict__ out, int B) {
    int   t   = blockIdx.x * blockDim.x + threadIdx.x;
    float val = 0.0f;
    if (t < B) {
        int ix = indices[t];
        int lc = labels[t];

        float ci  = ins_cnt[ix];                 // >= 1 (this sample counted)
        float mi  = ins_sum[ix] / ci;
        float ai  = __expf(ci * LN_SMOOTH);      // 0.9^cnt
        float sli = ai * mem_ins[ix] + (1.0f - ai) * mi;

        float cc  = cls_cnt[lc];
        float mc  = cls_sum[lc] / cc;
        float ac  = __expf(cc * LN_SMOOTH);
        float slc = ac * mem_cls[lc] + (1.0f - ac) * mc;

        val = loss[t] * optimal_conf(sli) * optimal_conf(slc) * (1.0f / (float)B);
    }
    for (int off = WAVE / 2; off > 0; off >>= 1)
        val += __shfl_xor(val, off, WAVE);
    if ((threadIdx.x & (WAVE - 1)) == 0) atomicAdd(out, val);
}

extern "C" void kernel_launch(void* const* d_in, const int* in_sizes, int n_in,
                              void* d_out, int out_size, void* d_ws, size_t ws_size,
                              hipStream_t stream) {
    const float* preds   = (const float*)d_in[0];
    const float* mem_ins = (const float*)d_in[1];
    const float* mem_cls = (const float*)d_in[2];
    const int*   labels  = (const int*)d_in[3];
    const int*   indices = (const int*)d_in[4];

    const int B    = in_sizes[3];   // 32768
    const int NCLS = in_sizes[2];   // 1000
    const int NSMP = in_sizes[1];   // 1000000

    float* ws      = (float*)d_ws;
    float* loss    = ws;                 // B
    float* ins_sum = loss + B;           // NSMP
    float* ins_cnt = ins_sum + NSMP;     // NSMP
    float* cls_sum = ins_cnt + NSMP;     // NCLS
    float* cls_cnt = cls_sum + NCLS;     // NCLS
    float* out     = (float*)d_out;

    const int threads = 256;
    int initN  = (B > NCLS) ? B : NCLS;
    int gInit  = (initN + threads - 1) / threads;
    int gElem  = (B + threads - 1) / threads;
    int gCE    = (B < 4096) ? B : 4096;  // grid-stride: 8 rows/block at B=32768

    dsl_init<<<gInit, threads, 0, stream>>>(indices, ins_sum, ins_cnt, cls_sum,
                                            cls_cnt, out, B, NCLS);

    size_t shmem = 2u * (size_t)NCLS * sizeof(float);
    bool use_lds = ((NCLS & 3) == 0) && (shmem <= 64u * 1024u);
    if (use_lds) {
        dsl_ce_lds<<<gCE, threads, shmem, stream>>>(preds, labels, loss, B, NCLS);
    } else {
        dsl_ce<<<gCE, threads, 0, stream>>>(preds, labels, loss, B, NCLS);
    }

    dsl_scatter<<<gElem, threads, 0, stream>>>(loss, labels, indices, ins_sum,
                                               ins_cnt, cls_sum, cls_cnt, B);
    dsl_finalize<<<gElem, threads, 0, stream>>>(loss, labels, indices, mem_ins,
                                                mem_cls, ins_sum, ins_cnt,
                                                cls_sum, cls_cnt, out, B);
}